// JPEGBase_240518169043
// MI455X (gfx1250) — compile-verified
//
#include <hip/hip_runtime.h>
#include <hip/hip_bf16.h>

// JPEG DCT->IDCT pipeline for MI455X (gfx1250).
// Memory-bound problem (~201MB traffic @ 23.3TB/s ≈ 8.6us floor), so the
// kernel is a single fused pass. The per-8x8-block 2D DCT/IDCT is mapped
// onto V_WMMA_F32_16X16X4_F32: a 16x16 tile holds four 8x8 blocks, and the
// block-diagonal DD = diag(D,D) applies D per block. Right-multiplies are
// rewritten as left-multiply + transpose: f(X)=(DD*X)^T, f(f(T))=DD*T*DD^T.
// Transposes / operand-layout conversions go through a 16x18 f32 LDS tile
// (per-wave in-order DS ops; 1-wave workgroups so barriers are NOPs).

typedef float v2f __attribute__((ext_vector_type(2)));
typedef float v8f __attribute__((ext_vector_type(8)));

#define LDS_STRIDE 18   // pad to even (8B-aligned float2 reads) and dodge bank conflicts
#define IMG_W 512
#define IMG_H 512

__device__ __forceinline__ float dcoef(int i, int j) {
    // D[i][j] of the 8x8 DCT basis
    if (i == 0) return 0.35355339059327373f;             // sqrt(2/8)/sqrt(2)
    return 0.5f * __cosf(0.39269908169872414f * (float)i * ((float)j + 0.5f));
}

__device__ __forceinline__ float ddmat(int r, int c) {
    // Block-diagonal 16x16: diag(D, D)
    return ((r >> 3) == (c >> 3)) ? dcoef(r & 7, c & 7) : 0.0f;
}

// One 16x16x16 f32 matmul as 4 accumulating 16x16x4 WMMAs.
__device__ __forceinline__ v8f mm16x16(const v2f A[4], const v2f B[4]) {
    v8f acc = {0.f, 0.f, 0.f, 0.f, 0.f, 0.f, 0.f, 0.f};
#pragma unroll
    for (int c = 0; c < 4; ++c) {
        acc = __builtin_amdgcn_wmma_f32_16x16x4_f32(
            /*neg_a=*/false, A[c], /*neg_b=*/false, B[c],
            /*c_mod=*/(short)0, acc, /*reuse_a=*/false, /*reuse_b=*/false);
    }
    return acc;
}

// C/D layout (VGPR v, lane: M = v + 8*hi, N = lo) -> LDS row-major (stride 18)
__device__ __forceinline__ void cd_to_lds(float* lds, const v8f& acc, int lo, int hi) {
#pragma unroll
    for (int v = 0; v < 8; ++v)
        lds[(v + 8 * hi) * LDS_STRIDE + lo] = acc[v];
}

// Read B-operand layout of the TRANSPOSE of what's in LDS:
// chunk c VGPR pair: B[c].x = X^T[4c+2hi][lo] = lds[lo][4c+2hi], .y = next K
__device__ __forceinline__ void lds_to_Bt(const float* lds, v2f B[4], int lo, int hi) {
#pragma unroll
    for (int c = 0; c < 4; ++c) {
        int k = 4 * c + 2 * hi;
        B[c].x = lds[lo * LDS_STRIDE + k];
        B[c].y = lds[lo * LDS_STRIDE + k + 1];
    }
}

// Full DCT->IDCT chain on one 16x16 tile: four stages of (left-mul, transpose).
// o[v] = result[v + 8*hi][lo]  (row layout, ready for coalesced store)
__device__ __forceinline__ void dct_idct(float* lds, const v2f aDD[4], const v2f aDDt[4],
                                         v2f B[4], float o[8], int lo, int hi) {
    v8f acc = mm16x16(aDD, B);                               // DD * X
    __syncthreads(); cd_to_lds(lds, acc, lo, hi); __syncthreads();
    lds_to_Bt(lds, B, lo, hi);                               // transpose
    acc = mm16x16(aDD, B);                                   // -> DD*T*DD^T (DCT)
    __syncthreads(); cd_to_lds(lds, acc, lo, hi); __syncthreads();
    lds_to_Bt(lds, B, lo, hi);
    acc = mm16x16(aDDt, B);                                  // DD^T * Y
    __syncthreads(); cd_to_lds(lds, acc, lo, hi); __syncthreads();
    lds_to_Bt(lds, B, lo, hi);
    acc = mm16x16(aDDt, B);                                  // -> DD^T*Y*DD (IDCT)
    __syncthreads(); cd_to_lds(lds, acc, lo, hi); __syncthreads();
#pragma unroll
    for (int v = 0; v < 8; ++v)                              // final transpose read
        o[v] = lds[lo * LDS_STRIDE + v + 8 * hi];
    __syncthreads();
}

__global__ void __launch_bounds__(32)
jpeg_dct_kernel(const float* __restrict__ in, float* __restrict__ out) {
    __shared__ float lds[16 * LDS_STRIDE];

    const int L  = threadIdx.x;
    const int lo = L & 15;
    const int hi = L >> 4;

    // one block = one 16x16 spatial tile of one image, all 3 channels
    const int tile = blockIdx.x;
    const int tx   = tile & 31;
    const int ty   = (tile >> 5) & 31;
    const int img  = tile >> 10;

    const size_t plane = (size_t)IMG_W * IMG_H;
    const size_t tbase = (size_t)(ty * 16) * IMG_W + (size_t)(tx * 16);
    const float* pr = in + ((size_t)img * 3 + 0) * plane + tbase;
    const float* pg = in + ((size_t)img * 3 + 1) * plane + tbase;
    const float* pb = in + ((size_t)img * 3 + 2) * plane + tbase;

    // A-operand constants for DD and DD^T (16x4 chunk layout):
    // lane L: M = lo; VGPR0 -> K = 4c+2hi, VGPR1 -> K = 4c+2hi+1
    v2f aDD[4], aDDt[4];
#pragma unroll
    for (int c = 0; c < 4; ++c) {
        int k = 4 * c + 2 * hi;
        aDD[c].x  = ddmat(lo, k);     aDD[c].y  = ddmat(lo, k + 1);
        aDDt[c].x = ddmat(k, lo);     aDDt[c].y = ddmat(k + 1, lo);
    }

    // Load RGB tile directly in B-operand row layout:
    // q = 2c+p -> row = 4c + 2*hi + p, col = lo  (coalesced across lanes)
    float rr[8], gg[8], bb[8];
#pragma unroll
    for (int q = 0; q < 8; ++q) {
        int row = 4 * (q >> 1) + 2 * hi + (q & 1);
        rr[q] = pr[row * IMG_W + lo];
        gg[q] = pg[row * IMG_W + lo];
        bb[q] = pb[row * IMG_W + lo];
    }

    // RGB -> YUV, *255, packed as B operands
    v2f By[4], Bu[4], Bv[4];
#pragma unroll
    for (int c = 0; c < 4; ++c) {
#pragma unroll
        for (int p = 0; p < 2; ++p) {
            int q = 2 * c + p;
            float r = rr[q], g = gg[q], b = bb[q];
            float y = 255.0f * ( 0.299f * r + 0.587f * g + 0.114f * b);
            float u = 255.0f * (-0.147f * r - 0.289f * g + 0.436f * b);
            float v = 255.0f * ( 0.615f * r - 0.515f * g - 0.100f * b);
            if (p == 0) { By[c].x = y; Bu[c].x = u; Bv[c].x = v; }
            else        { By[c].y = y; Bu[c].y = u; Bv[c].y = v; }
        }
    }

    float oy[8], ou[8], ov[8];
    dct_idct(lds, aDD, aDDt, By, oy, lo, hi);
    dct_idct(lds, aDD, aDDt, Bu, ou, lo, hi);
    dct_idct(lds, aDD, aDDt, Bv, ov, lo, hi);

    // YUV -> RGB, /255, coalesced stores
    float* qr = out + ((size_t)img * 3 + 0) * plane + tbase;
    float* qg = out + ((size_t)img * 3 + 1) * plane + tbase;
    float* qb = out + ((size_t)img * 3 + 2) * plane + tbase;
    const float inv255 = 1.0f / 255.0f;
#pragma unroll
    for (int v = 0; v < 8; ++v) {
        int row = v + 8 * hi;
        float y = oy[v], u = ou[v], w = ov[v];
        qr[row * IMG_W + lo] = (y + 1.140f * w) * inv255;
        qg[row * IMG_W + lo] = (y - 0.396f * u - 0.581f * w) * inv255;
        qb[row * IMG_W + lo] = (y + 2.029f * u) * inv255;
    }
}

extern "C" void kernel_launch(void* const* d_in, const int* in_sizes, int n_in,
                              void* d_out, int out_size, void* d_ws, size_t ws_size,
                              hipStream_t stream) {
    // inputs: d_in[0] = i_co (unused by reference), d_in[1] = i_en
    const float* i_en = (const float*)d_in[1];
    float* out = (float*)d_out;

    // 32 images * (512/16)^2 tiles = 32768 one-wave blocks
    dim3 grid(32 * 32 * 32);
    dim3 block(32);
    jpeg_dct_kernel<<<grid, block, 0, stream>>>(i_en, out);
}